// HierarchicalDiffPool_15822659518961
// MI455X (gfx1250) — compile-verified
//
#include <hip/hip_runtime.h>
#include <hip/hip_bf16.h>

// ---------------------------------------------------------------------------
// GATv2 hierarchical pool for gfx1250 (MI455X), wave32.
//   Phase 1: x_l = x@Wl^T, x_r = x@Wr^T via V_WMMA_F32_16X16X4_F32 (fp32 WMMA)
//   Phase 2: edge-parallel GATv2 (8-head pool + 1-head assign) with
//            segment-softmax via encoded atomicMax + atomicAdd
//   Phase 3: exact descending top-k (bitonic sort of 64-bit keys), gather rows
// ---------------------------------------------------------------------------

typedef __attribute__((ext_vector_type(2))) float v2f;
typedef __attribute__((ext_vector_type(8))) float v8f;

#define CIN 64
#define HEADS 8
#define NEG_SLOPE 0.2f

__device__ __forceinline__ unsigned enc_f32(float f) {
  // order-preserving map: float compare == unsigned compare
  unsigned b = __float_as_uint(f);
  return (b & 0x80000000u) ? ~b : (b | 0x80000000u);
}
__device__ __forceinline__ float dec_f32(unsigned u) {
  unsigned b = (u & 0x80000000u) ? (u ^ 0x80000000u) : ~u;
  return __uint_as_float(b);
}
// encode(-inf): bits(-inf)=0xFF800000 -> ~ = 0x007FFFFF
#define ENC_NEG_INF 0x007FFFFFu

// ---------------------------------------------------------------------------
// Kernel 0: initialize accumulators in workspace + zero the tail of d_out
// ---------------------------------------------------------------------------
__global__ void init_ws_kernel(float* __restrict__ xp, float* __restrict__ scores,
                               float* __restrict__ dsum_p, float* __restrict__ dsum_a,
                               unsigned* __restrict__ dmax_p, unsigned* __restrict__ dmax_a,
                               float* __restrict__ out, int N, int tail0, int out_size) {
  int i = blockIdx.x * blockDim.x + threadIdx.x;
  if (i < N * CIN) xp[i] = 0.0f;                 // x_pooled accumulator
  if (i < N * HEADS) { dsum_p[i] = 0.0f; dmax_p[i] = ENC_NEG_INF; }
  if (i < N) { scores[i] = 0.0f; dsum_a[i] = 0.0f; dmax_a[i] = ENC_NEG_INF; }
  if (i < out_size - tail0) out[tail0 + i] = 0.0f;  // batch_coarse region == 0
}

// ---------------------------------------------------------------------------
// Kernel 1: x_l = x @ Wl^T and x_r = x @ Wr^T using V_WMMA_F32_16X16X4_F32.
// Block = 128 threads (4 waves); each wave owns 16 rows and produces the full
// 16x64 output tile for BOTH weight matrices: 2 mats x 4 n-tiles x 16 k-steps
// = 128 WMMA ops per wave. Weights live in LDS; A fragments stream from HBM
// exactly once (8B/lane/k-step == the 16x64 x-tile, no redundancy).
//
// A (16x4 f32, 2 VGPRs): lane<16 -> M=lane, {K0,K1}; lane>=16 -> M=lane-16,{K2,K3}
// B (4x16  f32, 2 VGPRs): lane<16 -> K={0,1} row pair? mirrored: v0 holds
//   B[kb][n=lane%16], v1 holds B[kb+1][n], kb = 4*kk + 2*(lane>=16).
// C/D (16x16 f32, 8 VGPRs): vgpr v, lane<16 -> (M=v, N=lane); lane>=16 -> (M=v+8)
// ---------------------------------------------------------------------------
__global__ __launch_bounds__(128) void gemm_wmma_kernel(
    const float* __restrict__ x, const float* __restrict__ Wl,
    const float* __restrict__ Wr, float* __restrict__ xl, float* __restrict__ xr) {
  __shared__ float sW[2][CIN * CIN];
  for (int i = threadIdx.x; i < CIN * CIN; i += 128) {
    sW[0][i] = Wl[i];
    sW[1][i] = Wr[i];
  }
  __syncthreads();

  const int wave   = threadIdx.x >> 5;   // wave32
  const int lane   = threadIdx.x & 31;
  const int laneHi = lane >> 4;          // 0/1 -> K sub-pair select
  const int laneLo = lane & 15;
  const int r0     = blockIdx.x * 64 + wave * 16;

  v8f acc[2][4] = {};
#pragma unroll
  for (int kk = 0; kk < 16; ++kk) {
    const int kb = kk * 4 + laneHi * 2;
    const float* xp = x + (size_t)(r0 + laneLo) * CIN + kb;
    v2f a; a.x = xp[0]; a.y = xp[1];
#pragma unroll
    for (int m = 0; m < 2; ++m) {
#pragma unroll
      for (int nt = 0; nt < 4; ++nt) {
        const float* wp = &sW[m][(nt * 16 + laneLo) * CIN + kb];  // B[k][n]=W[n][k]
        v2f b; b.x = wp[0]; b.y = wp[1];
        acc[m][nt] = __builtin_amdgcn_wmma_f32_16x16x4_f32(
            /*neg_a=*/false, a, /*neg_b=*/false, b,
            /*c_mod=*/(short)0, acc[m][nt], /*reuse_a=*/false, /*reuse_b=*/false);
      }
    }
  }
#pragma unroll
  for (int m = 0; m < 2; ++m) {
    float* outp = m ? xr : xl;
#pragma unroll
    for (int nt = 0; nt < 4; ++nt)
#pragma unroll
      for (int v = 0; v < 8; ++v) {
        const int row = r0 + v + laneHi * 8;
        outp[(size_t)row * CIN + nt * 16 + laneLo] = acc[m][nt][v];
      }
  }
}

// ---------------------------------------------------------------------------
// Kernel 2: assign-head GEMVs: s_l[n]=dot(x[n],Wl_a), s_r[n]=dot(x[n],Wr_a)
// ---------------------------------------------------------------------------
__global__ void assign_proj_kernel(const float* __restrict__ x,
                                   const float* __restrict__ Wla,
                                   const float* __restrict__ Wra,
                                   float* __restrict__ sl, float* __restrict__ sr,
                                   int N) {
  int n = blockIdx.x * blockDim.x + threadIdx.x;
  if (n >= N) return;
  const float4* xv = (const float4*)(x + (size_t)n * CIN);
  const float4* wl = (const float4*)Wla;
  const float4* wr = (const float4*)Wra;
  float al = 0.0f, ar = 0.0f;
#pragma unroll
  for (int q = 0; q < CIN / 4; ++q) {
    float4 v = xv[q], a = wl[q], b = wr[q];
    al += v.x * a.x + v.y * a.y + v.z * a.z + v.w * a.w;
    ar += v.x * b.x + v.y * b.y + v.z * b.z + v.w * b.w;
  }
  sl[n] = al;
  sr[n] = ar;
}

// ---------------------------------------------------------------------------
// Kernel 3 (edge pass 1): logits + segment max (encoded u32 atomicMax)
// ---------------------------------------------------------------------------
__global__ void edge_logits_kernel(const long long* __restrict__ src,
                                   const long long* __restrict__ dst,
                                   const float* __restrict__ xl, const float* __restrict__ xr,
                                   const float* __restrict__ att,
                                   const float* __restrict__ sl, const float* __restrict__ sr,
                                   const float* __restrict__ att_a,
                                   float* __restrict__ lg, float* __restrict__ la,
                                   unsigned* __restrict__ dmax_p, unsigned* __restrict__ dmax_a,
                                   int N, int E) {
  int e = blockIdx.x * blockDim.x + threadIdx.x;
  if (e >= E) return;
  const long long s = src[e], d = dst[e];
  const float4* xj = (const float4*)(xl + (size_t)s * CIN);
  const float4* xi = (const float4*)(xr + (size_t)d * CIN);
  const float4* at = (const float4*)att;
#pragma unroll
  for (int h = 0; h < HEADS; ++h) {
    float acc = 0.0f;
#pragma unroll
    for (int q = 0; q < 2; ++q) {
      float4 a = xj[h * 2 + q], b = xi[h * 2 + q], w = at[h * 2 + q];
      float t;
      t = a.x + b.x; acc += (t > 0.0f ? t : NEG_SLOPE * t) * w.x;
      t = a.y + b.y; acc += (t > 0.0f ? t : NEG_SLOPE * t) * w.y;
      t = a.z + b.z; acc += (t > 0.0f ? t : NEG_SLOPE * t) * w.z;
      t = a.w + b.w; acc += (t > 0.0f ? t : NEG_SLOPE * t) * w.w;
    }
    lg[(size_t)h * E + e] = acc;
    atomicMax(&dmax_p[(size_t)h * N + d], enc_f32(acc));
  }
  float t = sl[s] + sr[d];
  t = (t > 0.0f ? t : NEG_SLOPE * t) * att_a[0];
  la[e] = t;
  atomicMax(&dmax_a[d], enc_f32(t));
}

// ---------------------------------------------------------------------------
// Kernel 4 (edge pass 2): exp(logit - max[dst]) + segment sum
// ---------------------------------------------------------------------------
__global__ void edge_exp_kernel(const long long* __restrict__ dst,
                                float* __restrict__ lg, float* __restrict__ la,
                                const unsigned* __restrict__ dmax_p,
                                const unsigned* __restrict__ dmax_a,
                                float* __restrict__ dsum_p, float* __restrict__ dsum_a,
                                int N, int E) {
  int e = blockIdx.x * blockDim.x + threadIdx.x;
  if (e >= E) return;
  const long long d = dst[e];
#pragma unroll
  for (int h = 0; h < HEADS; ++h) {
    float m = dec_f32(dmax_p[(size_t)h * N + d]);
    float ex = __expf(lg[(size_t)h * E + e] - m);
    lg[(size_t)h * E + e] = ex;
    atomicAdd(&dsum_p[(size_t)h * N + d], ex);
  }
  float ma = dec_f32(dmax_a[d]);
  float ea = __expf(la[e] - ma);
  la[e] = ea;
  atomicAdd(&dsum_a[d], ea);
}

// ---------------------------------------------------------------------------
// Kernel 5 (edge pass 3): alpha-weighted scatter of x_l[src] -> x_pooled[dst]
// and scalar assign score scatter.
// ---------------------------------------------------------------------------
__global__ void edge_scatter_kernel(const long long* __restrict__ src,
                                    const long long* __restrict__ dst,
                                    const float* __restrict__ xl,
                                    const float* __restrict__ lg, const float* __restrict__ la,
                                    const float* __restrict__ dsum_p,
                                    const float* __restrict__ dsum_a,
                                    const float* __restrict__ sl,
                                    float* __restrict__ xp, float* __restrict__ scores,
                                    int N, int E) {
  int e = blockIdx.x * blockDim.x + threadIdx.x;
  if (e >= E) return;
  const long long s = src[e], d = dst[e];
#pragma unroll
  for (int h = 0; h < HEADS; ++h) {
    float alpha = lg[(size_t)h * E + e] / (dsum_p[(size_t)h * N + d] + 1e-16f);
    const float4* xj = (const float4*)(xl + (size_t)s * CIN + h * 8);
    float* op = xp + (size_t)d * CIN + h * 8;
    float4 v0 = xj[0], v1 = xj[1];
    atomicAdd(op + 0, v0.x * alpha); atomicAdd(op + 1, v0.y * alpha);
    atomicAdd(op + 2, v0.z * alpha); atomicAdd(op + 3, v0.w * alpha);
    atomicAdd(op + 4, v1.x * alpha); atomicAdd(op + 5, v1.y * alpha);
    atomicAdd(op + 6, v1.z * alpha); atomicAdd(op + 7, v1.w * alpha);
  }
  float aa = la[e] / (dsum_a[d] + 1e-16f);
  atomicAdd(&scores[d], aa * sl[s]);
}

// ---------------------------------------------------------------------------
// Top-k: softmax is monotonic -> rank raw scores. Pack a 64-bit key so that
// sorting DESCENDING gives jax.lax.top_k order (higher score first; ties ->
// lower index first because ~index is larger for smaller index).
// ---------------------------------------------------------------------------
__global__ void make_keys_kernel(const float* __restrict__ scores,
                                 unsigned long long* __restrict__ keys, int N) {
  int n = blockIdx.x * blockDim.x + threadIdx.x;
  if (n >= N) return;
  keys[n] = ((unsigned long long)enc_f32(scores[n]) << 32) |
            (unsigned long long)(0xFFFFFFFFu - (unsigned)n);
}

__global__ void bitonic_step_kernel(unsigned long long* __restrict__ keys,
                                    int size, int stride) {
  int i = blockIdx.x * blockDim.x + threadIdx.x;  // 0 .. N/2-1
  int pos = 2 * i - (i & (stride - 1));
  int a = pos, b = pos + stride;
  bool desc = ((pos & size) == 0);
  unsigned long long ka = keys[a], kb = keys[b];
  if ((ka < kb) == desc) { keys[a] = kb; keys[b] = ka; }
}

// ---------------------------------------------------------------------------
// Kernel 7: gather the top-k pooled rows into d_out (row-major [k,64]).
// ---------------------------------------------------------------------------
__global__ void gather_out_kernel(const unsigned long long* __restrict__ keys,
                                  const float* __restrict__ xp,
                                  float* __restrict__ out, int K) {
  int t = blockIdx.x * blockDim.x + threadIdx.x;
  if (t >= K * CIN) return;
  int i = t >> 6, c = t & 63;
  unsigned n = 0xFFFFFFFFu - (unsigned)(keys[i] & 0xFFFFFFFFull);
  out[t] = xp[(size_t)n * CIN + c];
}

// ---------------------------------------------------------------------------
extern "C" void kernel_launch(void* const* d_in, const int* in_sizes, int n_in,
                              void* d_out, int out_size, void* d_ws, size_t ws_size,
                              hipStream_t stream) {
  const float*     x       = (const float*)d_in[0];
  const long long* ei      = (const long long*)d_in[1];   // int64 [2,E]
  const float*     Wl_pool = (const float*)d_in[3];
  const float*     Wr_pool = (const float*)d_in[4];
  const float*     attp    = (const float*)d_in[5];
  const float*     Wl_a    = (const float*)d_in[6];
  const float*     Wr_a    = (const float*)d_in[7];
  const float*     att_a   = (const float*)d_in[8];

  const int N = in_sizes[0] / CIN;   // 65536
  const int E = in_sizes[1] / 2;     // 1048576
  const int K = N / 4;               // 16384

  const long long* srcp = ei;
  const long long* dstp = ei + E;

  // workspace carve-up (all region sizes are multiples of 16B)
  float* xl     = (float*)d_ws;                       // N*64
  float* xr     = xl + (size_t)N * CIN;               // N*64
  float* xp     = xr + (size_t)N * CIN;               // N*64 (x_pooled accum)
  float* lg     = xp + (size_t)N * CIN;               // E*8  pool logits/exp
  float* la     = lg + (size_t)E * HEADS;             // E    assign logit/exp
  float* sl     = la + (size_t)E;                     // N
  float* sr     = sl + N;                             // N
  float* dsum_p = sr + N;                             // N*8
  float* dsum_a = dsum_p + (size_t)N * HEADS;         // N
  float* scores = dsum_a + N;                         // N
  unsigned* dmax_p = (unsigned*)(scores + N);         // N*8
  unsigned* dmax_a = dmax_p + (size_t)N * HEADS;      // N
  unsigned long long* keys = (unsigned long long*)(dmax_a + N);  // N (8B-aligned)

  float* out = (float*)d_out;
  (void)n_in; (void)ws_size;

  // 0) init accumulators + zero batch_coarse tail of d_out
  {
    int blocks = (N * CIN + 255) / 256;
    init_ws_kernel<<<blocks, 256, 0, stream>>>(xp, scores, dsum_p, dsum_a,
                                               dmax_p, dmax_a, out, N,
                                               K * CIN, out_size);
  }
  // 1) fp32 WMMA GEMMs
  gemm_wmma_kernel<<<N / 64, 128, 0, stream>>>(x, Wl_pool, Wr_pool, xl, xr);
  // 2) assign projections
  assign_proj_kernel<<<(N + 255) / 256, 256, 0, stream>>>(x, Wl_a, Wr_a, sl, sr, N);
  // 3-5) edge passes
  {
    int eb = (E + 255) / 256;
    edge_logits_kernel<<<eb, 256, 0, stream>>>(srcp, dstp, xl, xr, attp, sl, sr,
                                               att_a, lg, la, dmax_p, dmax_a, N, E);
    edge_exp_kernel<<<eb, 256, 0, stream>>>(dstp, lg, la, dmax_p, dmax_a,
                                            dsum_p, dsum_a, N, E);
    edge_scatter_kernel<<<eb, 256, 0, stream>>>(srcp, dstp, xl, lg, la,
                                                dsum_p, dsum_a, sl, xp, scores, N, E);
  }
  // 6) exact descending top-k via bitonic sort of 64-bit keys (L2-resident)
  make_keys_kernel<<<(N + 255) / 256, 256, 0, stream>>>(scores, keys, N);
  {
    int half = N / 2;
    int blocks = (half + 255) / 256;
    for (int size = 2; size <= N; size <<= 1)
      for (int stride = size >> 1; stride > 0; stride >>= 1)
        bitonic_step_kernel<<<blocks, 256, 0, stream>>>(keys, size, stride);
  }
  // 7) gather selected rows
  gather_out_kernel<<<(K * CIN + 255) / 256, 256, 0, stream>>>(keys, xp, out, K);
}